// PartsLayer_9363028705611
// MI455X (gfx1250) — compile-verified
//
#include <hip/hip_runtime.h>

typedef __attribute__((ext_vector_type(2))) float v2f;
typedef __attribute__((ext_vector_type(8))) float v8f;

// B=16384, P=64, K=64, O=64
// out[b,p,o] = relu( sum_k x[b,p,k]*W[p,k,o] + bias[p,o] )
//
// Wave-level plan (wave32, V_WMMA_F32_16X16X4_F32):
//   - A operand (16x4 f32): lanes 0-15 hold M=lane, K={0,1}; lanes 16-31 hold
//     M=lane-16, K={2,3}  -> per k-step each lane does one 8B load from x.
//   - B operand (4x16 f32): VGPR0 = K rows {4kk, 4kk+2} striped over lane halves,
//     VGPR1 = K rows {4kk+1, 4kk+3}; N = lane&15 (+16n per O-tile).
//   - C/D (16x16 f32): VGPR r holds row r (lanes 0-15) / row r+8 (lanes 16-31),
//     col = 16n + (lane&15).
// W[p] (16KB) is register-resident (64 v2f operands), reused over 8 B-tiles.

__global__ __launch_bounds__(256, 1)
void parts_block_diag_kernel(const float* __restrict__ x,
                             const float* __restrict__ W,
                             const float* __restrict__ bias,
                             float* __restrict__ out) {
    const int p    = blockIdx.x;          // part, 0..63
    const int wave = threadIdx.x >> 5;    // 0..7
    const int lane = threadIdx.x & 31;
    const int half = lane >> 4;           // 0: lanes 0-15, 1: lanes 16-31
    const int l    = lane & 15;

    // ---- Preload W[p] (64x64) into registers as 16x4 WMMA B-operands ----
    const float* Wp = W + p * 64 * 64;
    v2f wB[16][4];
#pragma unroll
    for (int kk = 0; kk < 16; ++kk) {
#pragma unroll
        for (int n = 0; n < 4; ++n) {
            const int row = 4 * kk + 2 * half;
            const int col = 16 * n + l;
            v2f b;
            b.x = Wp[row * 64 + col];        // VGPR0: K = 4kk + 2*half
            b.y = Wp[(row + 1) * 64 + col];  // VGPR1: K = 4kk + 2*half + 1
            wB[kk][n] = b;
        }
    }

    // ---- Bias for this part: per output column (same for both lane halves) ----
    float bv[4];
#pragma unroll
    for (int n = 0; n < 4; ++n) bv[n] = bias[p * 64 + 16 * n + l];

    // ---- 8 B-tiles of 16 rows per wave ----
    const int tile0 = (blockIdx.y * 8 + wave) * 8;   // gridDim.y = 16 -> 1024 tiles
    for (int t = 0; t < 8; ++t) {
        const int b0 = (tile0 + t) * 16;

        // A operands: lane loads 8B of row (b0 + l), k = 4kk + 2*half
        const float* xrow = x + ((b0 + l) * 64 + p) * 64 + 2 * half;
        v2f a[16];
#pragma unroll
        for (int kk = 0; kk < 16; ++kk)
            a[kk] = *(const v2f*)(xrow + 4 * kk);

        // Accumulators initialized with bias (bias is per-column -> splat over rows)
        v8f acc[4];
#pragma unroll
        for (int n = 0; n < 4; ++n) {
#pragma unroll
            for (int r = 0; r < 8; ++r) acc[n][r] = bv[n];
        }

        // 16 chained K-steps x 4 O-tiles = 64 WMMAs
#pragma unroll
        for (int kk = 0; kk < 16; ++kk) {
#pragma unroll
            for (int n = 0; n < 4; ++n) {
                acc[n] = __builtin_amdgcn_wmma_f32_16x16x4_f32(
                    /*neg_a=*/false, a[kk],
                    /*neg_b=*/false, wB[kk][n],
                    /*c_mod=*/(short)0, acc[n],
                    /*reuse_a=*/false, /*reuse_b=*/false);
            }
        }

        // ReLU + store: lane writes col 16n+l of row b0 + r + 8*half
#pragma unroll
        for (int n = 0; n < 4; ++n) {
#pragma unroll
            for (int r = 0; r < 8; ++r) {
                float v = acc[n][r];
                v = v > 0.0f ? v : 0.0f;
                out[((b0 + r + 8 * half) * 64 + p) * 64 + 16 * n + l] = v;
            }
        }
    }
}

extern "C" void kernel_launch(void* const* d_in, const int* in_sizes, int n_in,
                              void* d_out, int out_size, void* d_ws, size_t ws_size,
                              hipStream_t stream) {
    const float* x    = (const float*)d_in[0];  // [16384, 64, 64]
    const float* W    = (const float*)d_in[1];  // [64, 64, 64]
    const float* bias = (const float*)d_in[2];  // [64, 64]
    float* out        = (float*)d_out;          // [16384, 64, 64]

    // grid.x = P (64 parts), grid.y = 16 chunks; 8 waves/block * 8 tiles/wave
    // * 16 chunks = 1024 tiles of 16 rows = 16384 rows.
    dim3 grid(64, 16);
    dim3 block(256);
    parts_block_diag_kernel<<<grid, block, 0, stream>>>(x, W, bias, out);
}